// DomainQueue_36773509988901
// MI455X (gfx1250) — compile-verified
//
#include <hip/hip_runtime.h>
#include <math.h>

#define B_ 64
#define L_ 1024
#define C_ 512
#define NUM_DOMAINS 4
#define CAPACITY 1024
#define EPS 1e-6f
#define LSPLIT 4
#define LCHUNK (L_ / LSPLIT)   // 256 rows per split

typedef __attribute__((ext_vector_type(2))) float v2f;
typedef __attribute__((ext_vector_type(4))) float v4f;
typedef __attribute__((ext_vector_type(8))) float v8f;

// ---------------------------------------------------------------------------
// Kernel A: per-(b,c) partial sum / sum-of-squares over a chunk of L rows.
// One wave32 owns 32 channels as TWO 16-column WMMA tiles (even/odd channels)
// and reduces LCHUNK rows in K=4 tiles using V_WMMA_F32_16X16X4_F32 with an
// all-ones A matrix:  D[m,n] = sum_k B[k,n] + C[m,n]  (per-column sums).
// Each lane loads float2 (b64) covering (c0,c0+1) of one row, so one pair of
// loads feeds both tiles' v2f B operands:
//   tile0 (col n -> channel cbase+2n)   : { r0.x, r1.x }
//   tile1 (col n -> channel cbase+2n+1) : { r0.y, r1.y }
// With all-ones A, any K permutation of the B layout still yields the correct
// column sum; results are read from accumulator VGPR0 on lanes 0-15.
// ---------------------------------------------------------------------------
__global__ __launch_bounds__(128) void dq_stats_kernel(
    const float* __restrict__ x, float* __restrict__ psum,
    float* __restrict__ psq) {
  const int wave  = threadIdx.x >> 5;          // 0..3
  const int lane  = threadIdx.x & 31;
  const int b     = blockIdx.z;
  const int split = blockIdx.y;
  const int cbase = blockIdx.x * 128 + wave * 32;  // 32 channels per wave
  const int n     = lane & 15;                 // column within tile
  const int koff  = (lane >> 4) * 2;           // rows {0,1} or {2,3} of K-tile
  const int c0    = cbase + 2 * n;             // even channel of this lane

  const float* xp =
      x + ((size_t)b * L_ + (size_t)split * LCHUNK) * C_ + c0;

  v8f accS0 = {0.f, 0.f, 0.f, 0.f, 0.f, 0.f, 0.f, 0.f};
  v8f accS1 = accS0, accQ0 = accS0, accQ1 = accS0;
  const v2f ones = {1.0f, 1.0f};

  for (int l = 0; l < LCHUNK; l += 4) {
    const float* p = xp + (size_t)(l + koff) * C_;
    const v2f r0 = *(const v2f*)(p);        // row l+koff:   (c0, c0+1)
    const v2f r1 = *(const v2f*)(p + C_);   // row l+koff+1: (c0, c0+1)
    v2f b0, b1;
    b0.x = r0.x; b0.y = r1.x;   // even-channel tile, K pair
    b1.x = r0.y; b1.y = r1.y;   // odd-channel tile,  K pair
    v2f q0 = b0 * b0;
    v2f q1 = b1 * b1;
    accS0 = __builtin_amdgcn_wmma_f32_16x16x4_f32(
        false, ones, false, b0, (short)0, accS0, false, false);
    accS1 = __builtin_amdgcn_wmma_f32_16x16x4_f32(
        false, ones, false, b1, (short)0, accS1, false, false);
    accQ0 = __builtin_amdgcn_wmma_f32_16x16x4_f32(
        false, ones, false, q0, (short)0, accQ0, false, false);
    accQ1 = __builtin_amdgcn_wmma_f32_16x16x4_f32(
        false, ones, false, q1, (short)0, accQ1, false, false);
  }

  if (lane < 16) {  // D[0,n] lives in acc VGPR0 on lanes 0-15
    size_t o = ((size_t)b * LSPLIT + split) * C_ + c0;
    v2f s, q;
    s.x = accS0[0]; s.y = accS1[0];
    q.x = accQ0[0]; q.y = accQ1[0];
    *(v2f*)(psum + o) = s;
    *(v2f*)(psq + o)  = q;
  }
}

// ---------------------------------------------------------------------------
// Kernel B: per-sample queue logic + fold everything into scale/bias.
//   mu  = sum/L ;  var = (sq - sum^2/L)/(L-1) ;  sig = sqrt(var+EPS)
//   gather (mu1,sig1) from queue slot (d_ind1[b], f_ind1[b]) -- unless the
//   circular-buffer scatter of this batch just wrote that slot, in which case
//   the stats of the writing sample b' override the queue contents.
//   a = sig_mix/sig ; bias = mu_mix - mu*a
// ---------------------------------------------------------------------------
__global__ __launch_bounds__(256) void dq_mix_kernel(
    const float* __restrict__ psum, const float* __restrict__ psq,
    const int* __restrict__ domain, const float* __restrict__ mean_q,
    const float* __restrict__ sig_q, const float* __restrict__ lmda,
    const int* __restrict__ d_ind1, const int* __restrict__ f_ind1,
    const int* __restrict__ offsets, float* __restrict__ a_out,
    float* __restrict__ bias_out) {
  const int b = blockIdx.x;
  __shared__ int s_src;
  if (threadIdx.x == 0) {
    const int d1 = d_ind1[b];
    const int f1 = f_ind1[b];
    int src = -1;
    if (d1 < NUM_DOMAINS) {  // only domains 0..3 were scattered this batch
      const int off = offsets[d1];
      int rank = 0;
      for (int bp = 0; bp < B_; ++bp) {
        if (domain[bp] == d1) {
          int pos = (off + rank) % CAPACITY;
          if (pos == f1) src = bp;  // ranks unique -> at most one match
          ++rank;
        }
      }
    }
    s_src = src;
  }
  __syncthreads();

  const int   src = s_src;
  const int   d1  = d_ind1[b];
  const int   f1  = f_ind1[b];
  const float lam = lmda[b];

  for (int c = threadIdx.x; c < C_; c += blockDim.x) {
    float s = 0.f, q = 0.f;
#pragma unroll
    for (int k = 0; k < LSPLIT; ++k) {
      s += psum[((size_t)b * LSPLIT + k) * C_ + c];
      q += psq[((size_t)b * LSPLIT + k) * C_ + c];
    }
    const float mu  = s * (1.0f / L_);
    const float var = (q - s * s * (1.0f / L_)) * (1.0f / (L_ - 1));
    const float sig = sqrtf(var + EPS);

    float mu1, sig1;
    if (src >= 0) {  // this batch's scatter overwrote the gathered slot
      float s2 = 0.f, q2 = 0.f;
#pragma unroll
      for (int k = 0; k < LSPLIT; ++k) {
        s2 += psum[((size_t)src * LSPLIT + k) * C_ + c];
        q2 += psq[((size_t)src * LSPLIT + k) * C_ + c];
      }
      mu1 = s2 * (1.0f / L_);
      const float var2 = (q2 - s2 * s2 * (1.0f / L_)) * (1.0f / (L_ - 1));
      sig1 = sqrtf(var2 + EPS);
    } else {
      const size_t qo = ((size_t)d1 * CAPACITY + f1) * C_ + c;
      mu1  = mean_q[qo];
      sig1 = sig_q[qo];
    }

    const float mu_mix  = mu * lam + mu1 * (1.0f - lam);
    const float sig_mix = sig * lam + sig1 * (1.0f - lam);
    const float a       = sig_mix / sig;
    a_out[(size_t)b * C_ + c]    = a;
    bias_out[(size_t)b * C_ + c] = mu_mix - mu * a;
  }
}

// ---------------------------------------------------------------------------
// Kernel C: streaming out = x*a[b,c] + bias[b,c], v4f (b128) loads/stores.
// x (128 MB) fits in the 192 MB L2 and was just pulled in by the stats pass,
// so the x reads here should largely hit L2. The 128 MB of output is stored
// NON-TEMPORALLY (TH=NT) so it does not evict x from L2 on its way to HBM.
// ---------------------------------------------------------------------------
__global__ __launch_bounds__(256) void dq_apply_kernel(
    const float* __restrict__ x, const float* __restrict__ a_arr,
    const float* __restrict__ bias_arr, float* __restrict__ out) {
  const size_t idx = (size_t)blockIdx.x * blockDim.x + threadIdx.x;  // v4f idx
  const int C4 = C_ / 4;                       // 128 v4f per row
  const int c4 = (int)(idx & (C4 - 1));
  const size_t row = idx >> 7;                 // b*L + l
  const int b = (int)(row >> 10);              // row / L

  const v4f xv = ((const v4f*)x)[idx];
  const v4f av = ((const v4f*)a_arr)[(size_t)b * C4 + c4];
  const v4f bv = ((const v4f*)bias_arr)[(size_t)b * C4 + c4];
  v4f o;
  o.x = fmaf(xv.x, av.x, bv.x);
  o.y = fmaf(xv.y, av.y, bv.y);
  o.z = fmaf(xv.z, av.z, bv.z);
  o.w = fmaf(xv.w, av.w, bv.w);
  __builtin_nontemporal_store(o, &((v4f*)out)[idx]);
}

extern "C" void kernel_launch(void* const* d_in, const int* in_sizes, int n_in,
                              void* d_out, int out_size, void* d_ws,
                              size_t ws_size, hipStream_t stream) {
  const float* x       = (const float*)d_in[0];
  const int*   domain  = (const int*)d_in[1];
  const float* mean_q  = (const float*)d_in[2];
  const float* sig_q   = (const float*)d_in[3];
  const float* lmda    = (const float*)d_in[4];
  const int*   d_ind1  = (const int*)d_in[5];
  const int*   f_ind1  = (const int*)d_in[6];
  const int*   offsets = (const int*)d_in[7];
  float*       out     = (float*)d_out;

  float* psum  = (float*)d_ws;                 // [B, LSPLIT, C]
  float* psq   = psum + (size_t)B_ * LSPLIT * C_;
  float* a_arr = psq + (size_t)B_ * LSPLIT * C_;  // [B, C]
  float* b_arr = a_arr + (size_t)B_ * C_;         // [B, C]

  dim3 gA(C_ / 128, LSPLIT, B_);  // 4 waves/block, 32 channels/wave
  dq_stats_kernel<<<gA, 128, 0, stream>>>(x, psum, psq);

  dq_mix_kernel<<<B_, 256, 0, stream>>>(psum, psq, domain, mean_q, sig_q, lmda,
                                        d_ind1, f_ind1, offsets, a_arr, b_arr);

  const size_t n4 = (size_t)B_ * L_ * C_ / 4;  // 8,388,608 v4f
  dq_apply_kernel<<<(unsigned)(n4 / 256), 256, 0, stream>>>(x, a_arr, b_arr,
                                                            out);
}